// DeepSetsFunc_34196529611494
// MI455X (gfx1250) — compile-verified
//
#include <hip/hip_runtime.h>

typedef __attribute__((ext_vector_type(2))) float v2f;
typedef __attribute__((ext_vector_type(8))) float v8f;

#define L_DIM 256
#define S_DIM 64
#define D_DIM 128
#define H_DIM 512

// LDS row pitches padded by +4 floats: A-fragment ds_load_b64 walks 16 rows per
// half-wave; pitch % 64 == 4 spreads the 16 lanes across all 64 LDS banks.
#define XPITCH 132   // for 128-wide tiles (pitch mod 64 = 4)
#define HPITCH 516   // for 512-wide tile  (pitch mod 64 = 4)

// A fragment for V_WMMA_F32_16X16X4_F32 (16x4, f32):
// lane = m + 16*(k>>1), vgpr = k&1  ->  lane t reads A[t&15][kb + 2*(t>>4) .. +1]
__device__ __forceinline__ v2f lds_load_a(const float* sA, int pitch, int mbase,
                                          int kb, int lane) {
    const float* p = sA + (mbase + (lane & 15)) * pitch + kb + ((lane >> 4) << 1);
    return *(const v2f*)p;
}

// B fragment (4x16, f32), row-major W[K][N]:
// vgpr v holds rows k=v (lanes 0-15) and k=v+2 (lanes 16-31)
__device__ __forceinline__ v2f glb_load_b(const float* __restrict__ W, int n,
                                          int kb, int ncol, int lane) {
    const int k0 = kb + ((lane >> 4) << 1);
    v2f b;
    b.x = W[(size_t)(k0    ) * n + ncol];
    b.y = W[(size_t)(k0 + 1) * n + ncol];
    return b;
}

// One K-pass accumulating a full 64-row column stripe: 4 mtiles x 1 ntile.
// Per B-fragment (one L2 fetch) we issue 4 independent v_wmma on 4 accumulators.
__device__ __forceinline__ void gemm_k_pass(const float* sA, int pitch,
                                            const float* __restrict__ W, int n,
                                            int kcount, int kw0, int ncol,
                                            int lane, v8f acc[4]) {
    #pragma unroll 4
    for (int kb = 0; kb < kcount; kb += 4) {
        v2f b  = glb_load_b(W, n, kw0 + kb, ncol, lane);
        v2f a0 = lds_load_a(sA, pitch, 0,  kb, lane);
        v2f a1 = lds_load_a(sA, pitch, 16, kb, lane);
        v2f a2 = lds_load_a(sA, pitch, 32, kb, lane);
        v2f a3 = lds_load_a(sA, pitch, 48, kb, lane);
        acc[0] = __builtin_amdgcn_wmma_f32_16x16x4_f32(false, a0, false, b, (short)0, acc[0], false, false);
        acc[1] = __builtin_amdgcn_wmma_f32_16x16x4_f32(false, a1, false, b, (short)0, acc[1], false, false);
        acc[2] = __builtin_amdgcn_wmma_f32_16x16x4_f32(false, a2, false, b, (short)0, acc[2], false, false);
        acc[3] = __builtin_amdgcn_wmma_f32_16x16x4_f32(false, a3, false, b, (short)0, acc[3], false, false);
    }
}

__device__ __forceinline__ void store_stripe_lds(float* sD, int pitch, int ncol,
                                                 int kh, const v8f acc[4], bool relu) {
    #pragma unroll
    for (int mt = 0; mt < 4; ++mt) {
        #pragma unroll
        for (int v = 0; v < 8; ++v) {          // C/D: vgpr v -> rows v and v+8
            float x = acc[mt][v];
            sD[((mt << 4) + v + (kh << 3)) * pitch + ncol] = relu ? fmaxf(x, 0.0f) : x;
        }
    }
}

extern "C" __global__ void __launch_bounds__(256, 1)
deepsets_fused(const float* __restrict__ X,
               const float* __restrict__ W1, const float* __restrict__ B1,
               const float* __restrict__ W2, const float* __restrict__ B2,
               const float* __restrict__ W3, const float* __restrict__ B3,
               const float* __restrict__ W4, const float* __restrict__ B4,
               float* __restrict__ Out)
{
    extern __shared__ float smem[];
    float* sX = smem;                        // S_DIM x XPITCH : set_input tile
    float* sC = sX + S_DIM * XPITCH;         // S_DIM x XPITCH : combined, then cm
    float* sH = sC + S_DIM * XPITCH;         // S_DIM x HPITCH : hidden activations

    const int tid   = threadIdx.x;
    const int lane  = tid & 31;
    const int wave  = tid >> 5;              // 8 waves
    const int l     = blockIdx.x;
    const int nlane = lane & 15;
    const int kh    = lane >> 4;             // lane half (0/1)

    // ---- Stage 0: load set_input[l] (64x128) into LDS (float4, padded rows) ----
    {
        const float4* src = (const float4*)(X + (size_t)l * S_DIM * D_DIM);
        #pragma unroll
        for (int i = 0; i < 8; ++i) {
            const int f   = tid + (i << 8);      // float4 index 0..2047
            const int row = f >> 5;              // 32 float4 per row
            const int c4  = f & 31;
            float4 v = src[f];
            *(float4*)(sX + row * XPITCH + (c4 << 2)) = v;
        }
    }
    __syncthreads();

    // ---- Stage 1: masked all-pairs max == top-2 over set axis, clamp at 0 ----
    if (tid < D_DIM) {
        float m1 = -3.402823466e38f, m2 = -3.402823466e38f;
        int arg = -1;
        for (int i = 0; i < S_DIM; ++i) {
            const float v = sX[i * XPITCH + tid];
            if (v > m1)      { m2 = m1; m1 = v; arg = i; }
            else if (v > m2) { m2 = v; }
        }
        const float m1c = fmaxf(m1, 0.0f);     // diagonal term contributes 0
        const float m2c = fmaxf(m2, 0.0f);
        for (int j = 0; j < S_DIM; ++j) {
            sC[j * XPITCH + tid] = (j == arg) ? m2c : m1c;
        }
    }
    __syncthreads();

    // ---- GEMM1: sH = relu(sC @ W1 + b1)   [64,128]x[128,512] ----
    // N=512 -> 32 ntiles, 4 per wave; each wave covers all 4 mtiles (64 rows).
    {
        for (int nt = 0; nt < 4; ++nt) {
            const int ncol = (((wave << 2) + nt) << 4) + nlane;
            const float bb = B1[ncol];
            v8f acc[4] = { { bb, bb, bb, bb, bb, bb, bb, bb },
                           { bb, bb, bb, bb, bb, bb, bb, bb },
                           { bb, bb, bb, bb, bb, bb, bb, bb },
                           { bb, bb, bb, bb, bb, bb, bb, bb } };
            gemm_k_pass(sC, XPITCH, W1, H_DIM, D_DIM, 0, ncol, lane, acc);
            store_stripe_lds(sH, HPITCH, ncol, kh, acc, true);
        }
    }
    __syncthreads();

    // ---- GEMM2: sC = sH @ W2 + b2   [64,512]x[512,128]  (cm overwrites combined) ----
    // N=128 -> 8 ntiles, 1 per wave.
    {
        const int ncol = (wave << 4) + nlane;
        const float bb = B2[ncol];
        v8f acc[4] = { { bb, bb, bb, bb, bb, bb, bb, bb },
                       { bb, bb, bb, bb, bb, bb, bb, bb },
                       { bb, bb, bb, bb, bb, bb, bb, bb },
                       { bb, bb, bb, bb, bb, bb, bb, bb } };
        gemm_k_pass(sH, HPITCH, W2, D_DIM, H_DIM, 0, ncol, lane, acc);
        store_stripe_lds(sC, XPITCH, ncol, kh, acc, false);
    }
    __syncthreads();

    // ---- GEMM3: sH = relu(sX @ W3[0:128] + sC @ W3[128:256] + b3) ----
    {
        for (int nt = 0; nt < 4; ++nt) {
            const int ncol = (((wave << 2) + nt) << 4) + nlane;
            const float bb = B3[ncol];
            v8f acc[4] = { { bb, bb, bb, bb, bb, bb, bb, bb },
                           { bb, bb, bb, bb, bb, bb, bb, bb },
                           { bb, bb, bb, bb, bb, bb, bb, bb },
                           { bb, bb, bb, bb, bb, bb, bb, bb } };
            gemm_k_pass(sX, XPITCH, W3, H_DIM, D_DIM, 0,     ncol, lane, acc); // x-part
            gemm_k_pass(sC, XPITCH, W3, H_DIM, D_DIM, D_DIM, ncol, lane, acc); // cm-part
            store_stripe_lds(sH, HPITCH, ncol, kh, acc, true);
        }
    }
    __syncthreads();

    // ---- GEMM4 + residual: Out = sX + sH @ W4 + b4   [64,512]x[512,128] ----
    {
        const int ncol = (wave << 4) + nlane;
        const float bb = B4[ncol];
        v8f acc[4] = { { bb, bb, bb, bb, bb, bb, bb, bb },
                       { bb, bb, bb, bb, bb, bb, bb, bb },
                       { bb, bb, bb, bb, bb, bb, bb, bb },
                       { bb, bb, bb, bb, bb, bb, bb, bb } };
        gemm_k_pass(sH, HPITCH, W4, D_DIM, H_DIM, 0, ncol, lane, acc);

        float* __restrict__ outl = Out + (size_t)l * S_DIM * D_DIM;
        #pragma unroll
        for (int mt = 0; mt < 4; ++mt) {
            #pragma unroll
            for (int v = 0; v < 8; ++v) {
                const int row = (mt << 4) + v + (kh << 3);
                outl[row * D_DIM + ncol] = acc[mt][v] + sX[row * XPITCH + ncol];
            }
        }
    }
}

extern "C" void kernel_launch(void* const* d_in, const int* in_sizes, int n_in,
                              void* d_out, int out_size, void* d_ws, size_t ws_size,
                              hipStream_t stream) {
    (void)in_sizes; (void)n_in; (void)out_size; (void)d_ws; (void)ws_size;
    const float* X  = (const float*)d_in[0];
    const float* W1 = (const float*)d_in[1];
    const float* b1 = (const float*)d_in[2];
    const float* W2 = (const float*)d_in[3];
    const float* b2 = (const float*)d_in[4];
    const float* W3 = (const float*)d_in[5];
    const float* b3 = (const float*)d_in[6];
    const float* W4 = (const float*)d_in[7];
    const float* b4 = (const float*)d_in[8];
    float* Out = (float*)d_out;

    const size_t smem_bytes =
        (size_t)(2 * S_DIM * XPITCH + S_DIM * HPITCH) * sizeof(float); // ~195 KB (<320 KB/WGP)

    deepsets_fused<<<dim3(L_DIM), dim3(256), smem_bytes, stream>>>(
        X, W1, b1, W2, b2, W3, b3, W4, b4, Out);
}